// MKMMDLoss_57827439673814
// MI455X (gfx1250) — compile-verified
//
#include <hip/hip_runtime.h>
#include <hip/hip_bf16.h>

typedef __attribute__((ext_vector_type(2))) float v2f;
typedef __attribute__((ext_vector_type(8))) float v8f;

#define NN 4096
#define DD 256
#define MT 64                       // macro-tile edge (rows of X / rows of Y per block)
#define NMT (NN / MT)               // 64 macro-tiles per dim
#define TOT_ELEMS (16777216u)       // 4096*4096
#define MED_RANK (8388607u)         // (N*N - 1) / 2
#define PANEL_FLOATS (MT * DD)      // 16384 floats = 64 KB per panel
#define LDS_BYTES (2 * PANEL_FLOATS * 4)  // 128 KB

struct MedState {
    unsigned hist[256];   // radix histogram
    unsigned prefix;      // selected high bits of median d2 (as u32 pattern)
    unsigned krem;        // remaining rank within current filtered set
    unsigned pad[2];
    double   acc[3];      // 0: ss off-diag kernel sum, 1: st kernel sum, 2: tt off-diag
    float    cks[5];      // inv(2*bw_k^2)
};

// ---- CDNA5 async global->LDS copy (ASYNCcnt-tracked), 16B per lane ----------
__device__ __forceinline__ void async_b128(unsigned lds_byte_off, const float* gaddr) {
    // GV mode: LDS byte offset taken from the VDST VGPR, global address from the 64-bit VADDR
    asm volatile("global_load_async_to_lds_b128 %0, %1, off"
                 :: "v"(lds_byte_off), "v"(gaddr) : "memory");
}
__device__ __forceinline__ void wait_async0() {
    asm volatile("s_wait_asynccnt 0" ::: "memory");
}

// ---------------------------------------------------------------- init state
__global__ void k_init(MedState* __restrict__ st) {
    int t = threadIdx.x;
    st->hist[t] = 0u;
    if (t == 0) {
        st->prefix = 0u;
        st->krem   = MED_RANK;
        st->acc[0] = 0.0; st->acc[1] = 0.0; st->acc[2] = 0.0;
    }
}

// ---------------------------------------------------------------- row sqnorms
__global__ void k_sqnorm(const float* __restrict__ S, const float* __restrict__ T,
                         float* __restrict__ snorm, float* __restrict__ tnorm) {
    __shared__ float red[256];
    int row = blockIdx.x;  // 0..8191
    const float* src = (row < NN) ? (S + (size_t)row * DD)
                                  : (T + (size_t)(row - NN) * DD);
    float v = src[threadIdx.x];
    red[threadIdx.x] = v * v;
    __syncthreads();
    for (int s = 128; s > 0; s >>= 1) {
        if (threadIdx.x < s) red[threadIdx.x] += red[threadIdx.x + s];
        __syncthreads();
    }
    if (threadIdx.x == 0) {
        float* dst = (row < NN) ? snorm : tnorm;
        dst[row & (NN - 1)] = red[0];
    }
}

// ---------------- WMMA f32 macro-tile: 64x64 output per block, LDS-staged ----
// MODE 0: store clamped d2 to D2 buffer
// MODE 1: kernel-sum all elements into acc
// MODE 2: kernel-sum skipping diagonal (i==j) into acc
template <int MODE>
__global__ void k_tile(const float* __restrict__ X, const float* __restrict__ Y,
                       const float* __restrict__ xn, const float* __restrict__ yn,
                       float* __restrict__ D2, const float* __restrict__ cks,
                       double* __restrict__ acc) {
    extern __shared__ float smem[];
    float* Xs = smem;                    // [MT][DD]
    float* Ys = smem + PANEL_FLOATS;     // [MT][DD]

    const int t  = threadIdx.x;          // 0..255
    const int b  = blockIdx.x;           // 0..NMT*NMT-1
    const int bi = b >> 6;               // macro row
    const int bj = b & (NMT - 1);        // macro col

    const float* xpan = X + (size_t)bi * MT * DD;   // contiguous 64 KB panel
    const float* ypan = Y + (size_t)bj * MT * DD;

    // async stage both panels: 4096 chunks of 16B each per panel, 256 threads
    const unsigned xb = (unsigned)(size_t)(void*)Xs;   // LDS byte offset (low 32 bits)
    const unsigned yb = (unsigned)(size_t)(void*)Ys;
#pragma unroll
    for (int c = 0; c < PANEL_FLOATS / 4 / 256; c++) { // 16 iters
        const int idx = c * 256 + t;                   // 16B chunk index
        async_b128(xb + (unsigned)idx * 16u, xpan + (size_t)idx * 4);
        async_b128(yb + (unsigned)idx * 16u, ypan + (size_t)idx * 4);
    }
    wait_async0();
    __syncthreads();

    const int lane = t & 31;
    const int wave = t >> 5;
    const int row  = lane & 15;
    const int hi   = lane >> 4;
    const int khi  = hi * 2;             // lanes 16-31 hold K = k+2, k+3

    float k0 = 0.f, k1 = 0.f, k2 = 0.f, k3 = 0.f, k4 = 0.f;
    if (MODE != 0) { k0 = cks[0]; k1 = cks[1]; k2 = cks[2]; k3 = cks[3]; k4 = cks[4]; }

    float partial = 0.f;

#pragma unroll
    for (int s = 0; s < 2; s++) {        // 8 waves x 2 = 16 sub-tiles (4x4)
        const int tile = wave * 2 + s;
        const int tli  = tile >> 2;
        const int tlj  = tile & 3;

        const float* xr = Xs + (size_t)(tli * 16 + row) * DD + khi;  // A: M=row
        const float* yr = Ys + (size_t)(tlj * 16 + row) * DD + khi;  // B: N=row

        v8f cacc = {};
#pragma unroll 16
        for (int k = 0; k < DD; k += 4) {
            v2f a  = *(const v2f*)(xr + k);
            v2f bb = *(const v2f*)(yr + k);
            cacc = __builtin_amdgcn_wmma_f32_16x16x4_f32(false, a, false, bb,
                                                         (short)0, cacc, false, false);
        }

        const int i0 = bi * MT + tli * 16;
        const int j0 = bj * MT + tlj * 16;
        const float syn = yn[j0 + row];
#pragma unroll
        for (int r = 0; r < 8; r++) {
            const int m = i0 + r + 8 * hi;
            const int j = j0 + row;
            float d2 = xn[m] + syn - 2.0f * cacc[r];
            d2 = fmaxf(d2, 0.0f);
            if (MODE == 0) {
                D2[(size_t)m * NN + j] = d2;
            } else {
                if (MODE == 2 && m == j) continue;
                partial += __expf(-d2 * k0) + __expf(-d2 * k1) + __expf(-d2 * k2)
                         + __expf(-d2 * k3) + __expf(-d2 * k4);
            }
        }
    }

    if (MODE != 0) {
        partial *= 0.2f;
        for (int off = 16; off > 0; off >>= 1)
            partial += __shfl_down(partial, off, 32);
        if (lane == 0) atomicAdd(acc, (double)partial);
    }
}

// ------------------------------------------------- radix-select histogram pass
__global__ void k_hist(const unsigned* __restrict__ bits, MedState* __restrict__ st, int p) {
    __shared__ unsigned lh[256];
    lh[threadIdx.x] = 0u;
    __syncthreads();
    const unsigned prefix = st->prefix;
    const int shift = 32 - 8 * p;        // only used when p > 0
    const int dsh = 24 - 8 * p;
    const unsigned stride = gridDim.x * blockDim.x;
    for (unsigned i = blockIdx.x * blockDim.x + threadIdx.x; i < TOT_ELEMS; i += stride) {
        unsigned v = bits[i];
        bool ok = (p == 0) || ((v >> shift) == (prefix >> shift));
        if (ok) atomicAdd(&lh[(v >> dsh) & 0xFFu], 1u);
    }
    __syncthreads();
    if (lh[threadIdx.x]) atomicAdd(&st->hist[threadIdx.x], lh[threadIdx.x]);
}

// ------------------------------------------------- radix-select digit scan
__global__ void k_scan(MedState* __restrict__ st, int p) {
    unsigned k = st->krem;
    unsigned cum = 0;
    int d = 255;
    for (int i = 0; i < 256; i++) {
        unsigned h = st->hist[i];
        if (cum + h > k) { d = i; break; }
        cum += h;
    }
    st->prefix |= (unsigned)d << (24 - 8 * p);
    st->krem = k - cum;
    for (int i = 0; i < 256; i++) st->hist[i] = 0u;
}

// ------------------------------------------------- bandwidth constants
__global__ void k_bw(MedState* __restrict__ st) {
    float med2 = __uint_as_float(st->prefix);   // median of d2; bw_k^2 = med2 / 2^(k-2)
    float inv = 1.0f / (2.0f * med2);
    for (int k = 0; k < 5; k++)
        st->cks[k] = exp2f((float)(k - 2)) * inv;  // 1/(2*bw_k^2)
}

// ------------------------------------------------- SS kernel sum (stored d2)
__global__ void k_ss_sum(const float* __restrict__ D2, const MedState* __restrict__ st,
                         double* __restrict__ acc) {
    const float k0 = st->cks[0], k1 = st->cks[1], k2 = st->cks[2],
                k3 = st->cks[3], k4 = st->cks[4];
    const unsigned stride = gridDim.x * blockDim.x;
    float partial = 0.f;
    for (unsigned idx = blockIdx.x * blockDim.x + threadIdx.x; idx < TOT_ELEMS; idx += stride) {
        unsigned i = idx >> 12, j = idx & (NN - 1);
        if (i == j) continue;
        float d2 = D2[idx];
        partial += __expf(-d2 * k0) + __expf(-d2 * k1) + __expf(-d2 * k2)
                 + __expf(-d2 * k3) + __expf(-d2 * k4);
    }
    partial *= 0.2f;
    __shared__ float red[256];
    red[threadIdx.x] = partial;
    __syncthreads();
    for (int s = 128; s > 0; s >>= 1) {
        if (threadIdx.x < s) red[threadIdx.x] += red[threadIdx.x + s];
        __syncthreads();
    }
    if (threadIdx.x == 0) atomicAdd(acc, (double)red[0]);
}

// ------------------------------------------------- finalize scalar
__global__ void k_fin(const MedState* __restrict__ st, float* __restrict__ out) {
    const double n = (double)NN, m = (double)NN;
    double v = st->acc[0] / (n * (n - 1.0))
             + st->acc[2] / (m * (m - 1.0))
             - 2.0 * st->acc[1] / (n * m);
    out[0] = (float)v;
}

extern "C" void kernel_launch(void* const* d_in, const int* in_sizes, int n_in,
                              void* d_out, int out_size, void* d_ws, size_t ws_size,
                              hipStream_t stream) {
    const float* S = (const float*)d_in[0];
    const float* T = (const float*)d_in[1];
    float* out = (float*)d_out;

    char* ws = (char*)d_ws;
    const size_t D2_BYTES = (size_t)NN * NN * sizeof(float);   // 64 MB
    float*    D2    = (float*)ws;
    MedState* st    = (MedState*)(ws + D2_BYTES);
    float*    snorm = (float*)(ws + D2_BYTES + 4096);
    float*    tnorm = snorm + NN;

    const int gemm_blocks = NMT * NMT;   // 4096 blocks, 64x64 output each

    // 1. init state
    k_init<<<1, 256, 0, stream>>>(st);
    // 2. row squared norms (8192 rows: 4096 source + 4096 target)
    k_sqnorm<<<2 * NN, 256, 0, stream>>>(S, T, snorm, tnorm);
    // 3. d2_ss via LDS-staged WMMA macro-tiles, stored for median + SS sum
    k_tile<0><<<gemm_blocks, 256, LDS_BYTES, stream>>>(S, S, snorm, snorm, D2, nullptr, nullptr);
    // 4. 4-pass radix select of rank (N^2-1)/2 over d2_ss bit patterns
    for (int p = 0; p < 4; p++) {
        k_hist<<<1024, 256, 0, stream>>>((const unsigned*)D2, st, p);
        k_scan<<<1, 1, 0, stream>>>(st, p);
    }
    // 5. bandwidth constants 1/(2*bw_k^2)
    k_bw<<<1, 1, 0, stream>>>(st);
    // 6. SS kernel sum (off-diagonal) from stored d2
    k_ss_sum<<<4096, 256, 0, stream>>>(D2, st, &st->acc[0]);
    // 7. ST kernel sum, fused GEMM + RBF (all elements)
    k_tile<1><<<gemm_blocks, 256, LDS_BYTES, stream>>>(S, T, snorm, tnorm, nullptr, st->cks, &st->acc[1]);
    // 8. TT kernel sum, fused GEMM + RBF (off-diagonal)
    k_tile<2><<<gemm_blocks, 256, LDS_BYTES, stream>>>(T, T, tnorm, tnorm, nullptr, st->cks, &st->acc[2]);
    // 9. combine
    k_fin<<<1, 1, 0, stream>>>(st, out);
}